// GraphAttentionLayer_89739046683220
// MI455X (gfx1250) — compile-verified
//
#include <hip/hip_runtime.h>
#include <hip/hip_bf16.h>
#include <math.h>

// GraphAttentionLayer for MI455X (gfx1250, wave32, WMMA, async global->LDS).
// B=8, N=2048, F=U=128.
//
// Math: softmax_j(c_i + a_j + bs  masked) == m_ij*e_j / sum_j m_ij*e_j with
// e_j = exp(a_j - amax_b)  (c_i, bs cancel; masked exp underflows to exact 0).
// So:  out[b,i,:] = (sum_j m_ij e_j T[b,j,:]) / (sum_j m_ij e_j).

#define GA_B 8
#define GA_N 2048
#define GA_F 128
#define GA_U 128
#define GA_JC 32                    // T rows staged per LDS chunk (16 KB)
#define GA_NCHUNK (GA_N / GA_JC)    // 64

typedef __attribute__((ext_vector_type(2))) float v2f;
typedef __attribute__((ext_vector_type(8))) float v8f;

// Low 32 bits of a flat pointer into the LDS aperture == the LDS byte offset.
__device__ __forceinline__ uint32_t lds_offset(const void* p) {
  return (uint32_t)(uintptr_t)p;
}

// Async copy 16B/lane global -> LDS (tracked by ASYNCcnt).
__device__ __forceinline__ void async_g2l_b128(uint32_t lds_off, const void* gaddr) {
  asm volatile("global_load_async_to_lds_b128 %0, %1, off"
               :: "v"(lds_off), "v"((unsigned long long)(uintptr_t)gaddr)
               : "memory");
}

// ---------------------------------------------------------------------------
// Kernel 1: T[b,n,u] = sum_f X[b,n,f] * Wf[f,u] + bf[u]
// One wave per 16x16 tile of T. 8 waves / block.
// A-frag (16x4 f32): lane r (0..15) row=r holds K = {0,1}; lanes 16..31 K={2,3}.
// B-frag (4x16 f32): VGPR0 = rows {0 | 2} by lane half, VGPR1 = rows {1 | 3}.
// C/D (16x16 f32):  VGPR v, lane l: M = v + 8*(l>=16), Nc = l&15.
// ---------------------------------------------------------------------------
__global__ __launch_bounds__(256) void ga_gemm_T(const float* __restrict__ X,
                                                 const float* __restrict__ Wf,
                                                 const float* __restrict__ bf,
                                                 float* __restrict__ T) {
  const int lane  = threadIdx.x & 31;
  const int wave  = threadIdx.x >> 5;
  const int tile  = blockIdx.x * 8 + wave;           // 8192 tiles total
  const int b     = tile >> 10;                      // 1024 tiles per batch
  const int r     = tile & 1023;
  const int i0    = (r >> 3) * 16;                   // row tile (N/16 = 128)
  const int u0    = (r & 7) * 16;                    // col tile (U/16 = 8)

  const int mrow  = lane & 15;
  const int khalf = lane >> 4;                       // 0 or 1
  const int col   = u0 + mrow;

  const float* Xrow = X + ((size_t)b * GA_N + (i0 + mrow)) * GA_F;

  v8f acc = {};
#pragma unroll 8
  for (int k0 = 0; k0 < GA_F; k0 += 4) {
    const int kk = k0 + 2 * khalf;
    const v2f a = *reinterpret_cast<const v2f*>(Xrow + kk);     // X[row, kk..kk+1]
    v2f bm;
    bm.x = Wf[(size_t)kk * GA_U + col];                         // Wf[kk,   col]
    bm.y = Wf[(size_t)(kk + 1) * GA_U + col];                   // Wf[kk+1, col]
    acc = __builtin_amdgcn_wmma_f32_16x16x4_f32(
        false, a, false, bm, (short)0, acc, false, false);
  }

  const float bias = bf[col];
  float* Tout = T + ((size_t)b * GA_N + i0) * GA_U + col;
#pragma unroll
  for (int v = 0; v < 8; ++v) {
    const int M = v + 8 * khalf;
    Tout[(size_t)M * GA_U] = acc[v] + bias;
  }
}

// ---------------------------------------------------------------------------
// Kernel 2: a[b,n] = T[b,n,:] . Ws[0:128];  e[b,n] = exp(a - max_n a)
// One workgroup (256 threads) per batch.
// ---------------------------------------------------------------------------
__global__ __launch_bounds__(256) void ga_compute_e(const float* __restrict__ T,
                                                    const float* __restrict__ Ws,
                                                    float* __restrict__ e) {
  __shared__ float w1s[GA_U];
  __shared__ float ash[GA_N];
  __shared__ float red[256];

  const int b = blockIdx.x;
  const int t = threadIdx.x;

  if (t < GA_U) w1s[t] = Ws[t];                      // w1 = Ws[:U]
  __syncthreads();

  float localMax = -INFINITY;
  for (int n = t; n < GA_N; n += 256) {
    const float* Trow = T + ((size_t)b * GA_N + n) * GA_U;
    float d = 0.f;
#pragma unroll
    for (int k = 0; k < GA_U; k += 4) {
      const float4 x = *reinterpret_cast<const float4*>(Trow + k);
      d += x.x * w1s[k] + x.y * w1s[k + 1] + x.z * w1s[k + 2] + x.w * w1s[k + 3];
    }
    ash[n] = d;
    localMax = fmaxf(localMax, d);
  }
  red[t] = localMax;
  __syncthreads();
  for (int s = 128; s > 0; s >>= 1) {
    if (t < s) red[t] = fmaxf(red[t], red[t + s]);
    __syncthreads();
  }
  const float amax = red[0];

  for (int n = t; n < GA_N; n += 256)
    e[(size_t)b * GA_N + n] = expf(ash[n] - amax);
}

// ---------------------------------------------------------------------------
// Kernel 3: out[b,i,:] = (sum_j m_ij * e_j * T[b,j,:]) / (sum_j m_ij * e_j)
// One wave per (b, 16-row tile); 8 waves/block all on the same batch share
// double-buffered LDS tiles of T streamed by the async global->LDS engine.
// adj is streamed from HBM exactly once (the 134 MB roofline floor).
// ---------------------------------------------------------------------------
__global__ __launch_bounds__(256) void ga_attn(const int* __restrict__ adj,
                                               const float* __restrict__ T,
                                               const float* __restrict__ e,
                                               float* __restrict__ out) {
  __shared__ float Tsh[2][GA_JC * GA_U];   // 2 x 16 KB double buffer
  __shared__ float esh[GA_N];              // 8 KB: e[b,:]

  const int lane  = threadIdx.x & 31;
  const int wave  = threadIdx.x >> 5;
  const int b     = blockIdx.x >> 4;                 // 16 blocks per batch
  const int i0    = ((blockIdx.x & 15) * 8 + wave) * 16;

  const int mrow  = lane & 15;
  const int khalf = lane >> 4;

  const int*   adjRow = adj + ((size_t)b * GA_N + (i0 + mrow)) * GA_N;
  const float* Tb     = T + (size_t)b * GA_N * GA_U;
  const float* eb     = e + (size_t)b * GA_N;

  // Stage e[b,:] (8 KB): per wave a 1 KB slice = 2 async b128 ops.
  {
    const uint32_t ebase = lds_offset(&esh[0]);
#pragma unroll
    for (int q = 0; q < 2; ++q) {
      const uint32_t off = (uint32_t)(wave * 1024 + q * 512 + lane * 16);
      async_g2l_b128(ebase + off, (const char*)eb + off);
    }
  }
  // Stage a 16 KB chunk of T rows: per wave a 2 KB slice = 4 async b128 ops.
  auto stage = [&](int chunk, int buf) {
    const uint32_t lbase = lds_offset(&Tsh[buf][0]);
    const char* gp = (const char*)(Tb + (size_t)chunk * GA_JC * GA_U);
#pragma unroll
    for (int q = 0; q < 4; ++q) {
      const uint32_t off = (uint32_t)(wave * 2048 + q * 512 + lane * 16);
      async_g2l_b128(lbase + off, gp + off);
    }
  };
  stage(0, 0);

  v8f acc[8];
#pragma unroll
  for (int t = 0; t < 8; ++t) acc[t] = (v8f){};
  float sAcc = 0.f;                                  // partial row-sum of weights

  for (int c = 0; c < GA_NCHUNK; ++c) {
    // Prefetch next chunk (wraps to chunk 0 on the last iteration; it lands in
    // the dead buffer and simply retires before s_endpgm).
    stage((c + 1) & (GA_NCHUNK - 1), (c + 1) & 1);
    // 4 newest async ops may remain in flight; in-order retirement guarantees
    // everything for the current buffer (and, on c==0, esh) has landed.
    asm volatile("s_wait_asynccnt 4" ::: "memory");
    __syncthreads();

    const float* Ts = &Tsh[c & 1][0];
    const int j0 = c * GA_JC;
#pragma unroll
    for (int jj = 0; jj < GA_JC; jj += 4) {
      const int jl = jj + 2 * khalf;                 // local row in chunk
      const int jg = j0 + jl;                        // global j
      const int2 m2 = *reinterpret_cast<const int2*>(adjRow + jg);
      const v2f  e2 = *reinterpret_cast<const v2f*>(&esh[jg]);
      v2f a;
      a.x = (m2.x > 0) ? e2.x : 0.f;
      a.y = (m2.y > 0) ? e2.y : 0.f;
      sAcc += a.x + a.y;

      const float* r0 = Ts + jl * GA_U;              // T[b, j0+jl,   :] in LDS
      const float* r1 = r0 + GA_U;                   // T[b, j0+jl+1, :] in LDS
#pragma unroll
      for (int t = 0; t < 8; ++t) {
        v2f bm;
        bm.x = r0[t * 16 + mrow];
        bm.y = r1[t * 16 + mrow];
        acc[t] = __builtin_amdgcn_wmma_f32_16x16x4_f32(
            false, a, false, bm, (short)0, acc[t], false, false);
      }
    }
    __syncthreads();   // protect buffer reuse (chunk c+2 overwrites buf c&1)
  }

  // Combine the two lane-halves of sAcc: row r's weight sum lives in lanes r, r+16.
  const float Scomb = sAcc + __shfl_xor(sAcc, 16, 32);  // lane r (0..15) = S[row r]

  float* Ob = out + ((size_t)b * GA_N + i0) * GA_U + mrow;
#pragma unroll
  for (int v = 0; v < 8; ++v) {
    const int M = v + 8 * khalf;
    const float Sv  = __shfl(Scomb, M, 32);
    const float inv = (Sv > 0.f) ? (1.f / Sv) : 0.f; // all-masked row -> 0 (never hit w/ random adj)
#pragma unroll
    for (int t = 0; t < 8; ++t) {
      Ob[(size_t)M * GA_U + t * 16] = acc[t][v] * inv;
    }
  }
}

// ---------------------------------------------------------------------------
extern "C" void kernel_launch(void* const* d_in, const int* in_sizes, int n_in,
                              void* d_out, int out_size, void* d_ws, size_t ws_size,
                              hipStream_t stream) {
  const float* X   = (const float*)d_in[0];   // (B, N, F)
  const int*   adj = (const int*)d_in[1];     // (B, N, N)
  const float* Wf  = (const float*)d_in[2];   // (F, U)
  const float* bf  = (const float*)d_in[3];   // (U,)
  const float* Ws  = (const float*)d_in[4];   // (2U,)  -- only first U used (c,bs cancel)
  // d_in[5] = bs : cancels in softmax, unused.

  float* T = (float*)d_ws;                                   // 8 MB
  float* e = (float*)((char*)d_ws + (size_t)GA_B * GA_N * GA_U * sizeof(float));
  float* O = (float*)d_out;                                  // (B, N, U)

  // Kernel 1: 8192 tiles, 8 waves/block -> 1024 blocks.
  ga_gemm_T<<<1024, 256, 0, stream>>>(X, Wf, bf, T);
  // Kernel 2: one block per batch.
  ga_compute_e<<<GA_B, 256, 0, stream>>>(T, Ws, e);
  // Kernel 3: 1024 tiles, 8 waves/block -> 128 blocks.
  ga_attn<<<128, 256, 0, stream>>>(adj, T, e, O);
}